// GCNConvLFR_66829691125868
// MI455X (gfx1250) — compile-verified
//
#include <hip/hip_runtime.h>
#include <hip/hip_bf16.h>

#define IN_F   128
#define OUT_F  128
#define WPAD   132   // padded transposed-row stride: keeps ds_load_b64 8B-aligned
                     // and gives bank = (4*col + k) mod 64 -> conflict-free reads

typedef __attribute__((ext_vector_type(2))) float v2f;
typedef __attribute__((ext_vector_type(8))) float v8f;

// ---------------------------------------------------------------------------
// Kernel 1: support = X @ W  via V_WMMA_F32_16X16X4_F32 (full fp32 precision)
// Block = 128 threads = 4 waves. W staged TRANSPOSED in LDS: Wlds[col][k],
// so each B fragment (W[k][col], W[k+1][col]) is one aligned ds_load_b64.
// Each wave computes a 16-row x 128-col strip (8 16x16 tiles); the A fragment
// is loaded once per K-step and reused for all 8 B tiles.
// ---------------------------------------------------------------------------
__global__ __launch_bounds__(128)
void gcn_gemm_wmma(const float* __restrict__ X,
                   const float* __restrict__ W,
                   float* __restrict__ support,
                   int n_nodes) {
    __shared__ float Wlds[OUT_F * WPAD];   // 128*132*4 = 66 KB (WGP has 320 KB)

    const int tid = threadIdx.x;
    // Cooperative transposed load: Wlds[col*WPAD + k] = W[k*OUT_F + col]
    for (int i = tid; i < IN_F * OUT_F; i += 128) {
        const int k   = i >> 7;        // i / OUT_F
        const int col = i & (OUT_F - 1);
        Wlds[col * WPAD + k] = W[i];
    }
    __syncthreads();

    const int wave  = tid >> 5;
    const int lane  = tid & 31;
    const int mTile = blockIdx.x * 4 + wave;
    const int mBase = mTile * 16;
    if (mBase >= n_nodes) return;

    // A-matrix 16x4 layout (ISA 7.12.2): lanes 0-15 -> M=lane, K pair {0,1};
    // lanes 16-31 -> M=lane-16, K pair {2,3}.
    const int m16  = lane & 15;
    const int row  = min(mBase + m16, n_nodes - 1);   // clamp (tail safety)
    const int kOff = (lane >> 4) * 2;                 // 0 or 2

    v8f acc[8];
#pragma unroll
    for (int t = 0; t < 8; ++t) acc[t] = (v8f)0.0f;

    const float* xrow = X + (long)row * IN_F;

#pragma unroll 4
    for (int k0 = 0; k0 < IN_F; k0 += 4) {
        v2f a;
        a.x = xrow[k0 + kOff + 0];
        a.y = xrow[k0 + kOff + 1];
#pragma unroll
        for (int t = 0; t < 8; ++t) {
            // B-matrix 4x16: lanes 0-15 -> N=lane, K pair {0,1};
            // lanes 16-31 -> N=lane-16, K pair {2,3}.
            const int col = t * 16 + m16;
            // contiguous K-pair in transposed LDS -> one ds_load_b64
            v2f b = *(const v2f*)&Wlds[col * WPAD + k0 + kOff];
            acc[t] = __builtin_amdgcn_wmma_f32_16x16x4_f32(
                /*neg_a=*/false, a, /*neg_b=*/false, b,
                /*c_mod=*/(short)0, acc[t],
                /*reuse_a=*/false, /*reuse_b=*/false);
        }
    }

    // C/D 16x16 f32 layout: VGPR i -> M = i (lanes 0-15) or M = 8+i (lanes 16-31),
    // N = (lane & 15) within the tile.
    const int mStore = mBase + (lane >> 4) * 8;
#pragma unroll
    for (int t = 0; t < 8; ++t) {
        const int col = t * 16 + m16;
#pragma unroll
        for (int i = 0; i < 8; ++i) {
            const int r = mStore + i;
            if (r < n_nodes) {
                support[(long)r * OUT_F + col] = acc[t][i];
            }
        }
    }
}

// ---------------------------------------------------------------------------
// Kernel 2: out[n, f] = bias[f]   (d_out is poisoned; must initialize)
// ---------------------------------------------------------------------------
__global__ __launch_bounds__(256)
void gcn_bias_init(float* __restrict__ out,
                   const float* __restrict__ bias,
                   int total) {
    int i = blockIdx.x * blockDim.x + threadIdx.x;
    if (i < total) {
        out[i] = bias[i & (OUT_F - 1)];
    }
}

// ---------------------------------------------------------------------------
// Kernel 3: COO SpMM scatter. One wave (32 lanes) per edge; each lane handles
// a float4 (B128 gather; support is L2-resident at 25.6 MB vs 192 MB L2) and
// issues 4 hardware fp32 atomic adds. unsafeAtomicAdd guarantees the native
// global_atomic_add_f32 (no CAS loop); return value discarded -> non-returning
// variant tracked on STOREcnt.
// ---------------------------------------------------------------------------
__global__ __launch_bounds__(256)
void gcn_spmm_scatter(const int*   __restrict__ edge_row,
                      const int*   __restrict__ edge_col,
                      const float* __restrict__ edge_val,
                      const float* __restrict__ support,
                      float*       __restrict__ out,
                      int n_edges) {
    const int gwave = (blockIdx.x * blockDim.x + threadIdx.x) >> 5;
    const int lane  = threadIdx.x & 31;
    if (gwave >= n_edges) return;

    const int   src = edge_col[gwave];
    const int   dst = edge_row[gwave];
    const float val = edge_val[gwave];

    const float4* srow = (const float4*)(support + (long)src * OUT_F);
    float*        drow = out + (long)dst * OUT_F;

    float4 v = srow[lane];            // 32 lanes x 4 = 128 floats, coalesced
    const int f = lane * 4;
    unsafeAtomicAdd(drow + f + 0, val * v.x);
    unsafeAtomicAdd(drow + f + 1, val * v.y);
    unsafeAtomicAdd(drow + f + 2, val * v.z);
    unsafeAtomicAdd(drow + f + 3, val * v.w);
}

// ---------------------------------------------------------------------------
// Launcher
// inputs: x[50000*128] f32, edge_row[800000] i32, edge_col[800000] i32,
//         edge_val[800000] f32, weight[128*128] f32, bias[128] f32
// out: [50000*128] f32.  d_ws holds support (25.6 MB).
// ---------------------------------------------------------------------------
extern "C" void kernel_launch(void* const* d_in, const int* in_sizes, int n_in,
                              void* d_out, int out_size, void* d_ws, size_t ws_size,
                              hipStream_t stream) {
    const float* x        = (const float*)d_in[0];
    const int*   edge_row = (const int*)  d_in[1];
    const int*   edge_col = (const int*)  d_in[2];
    const float* edge_val = (const float*)d_in[3];
    const float* weight   = (const float*)d_in[4];
    const float* bias     = (const float*)d_in[5];
    float*       out      = (float*)d_out;
    float*       support  = (float*)d_ws;

    const int n_nodes = in_sizes[0] / IN_F;
    const int n_edges = in_sizes[1];

    // 1) support = X @ W (WMMA)
    const int mTiles = (n_nodes + 15) / 16;
    const int blocks = (mTiles + 3) / 4;          // 4 waves per block
    gcn_gemm_wmma<<<blocks, 128, 0, stream>>>(x, weight, support, n_nodes);

    // 2) out = bias (broadcast init)
    const int total = n_nodes * OUT_F;
    gcn_bias_init<<<(total + 255) / 256, 256, 0, stream>>>(out, bias, total);

    // 3) scatter-add edges
    const long threads = (long)n_edges * 32;
    gcn_spmm_scatter<<<(int)((threads + 255) / 256), 256, 0, stream>>>(
        edge_row, edge_col, edge_val, support, out, n_edges);
}